// ForwardSumLoss_77378130805413
// MI455X (gfx1250) — compile-verified
//
#include <hip/hip_runtime.h>
#include <hip/hip_bf16.h>
#include <stdint.h>

#ifndef __has_builtin
#define __has_builtin(x) 0
#endif

#define TKC 512                 // key length (threads per block, one per label)
#define NEGV (-1000000000.0f)
#define BLANK_LP (-1.0f)

#define AS1 __attribute__((address_space(1)))
#define AS3 __attribute__((address_space(3)))

// ---- CDNA5 async global->LDS copy (ASYNCcnt-tracked), builtin or inline asm ----
__device__ __forceinline__ void async_row_load(const float* g, float* l) {
#if __has_builtin(__builtin_amdgcn_global_load_async_to_lds_b32)
  __builtin_amdgcn_global_load_async_to_lds_b32((AS1 int*)g, (AS3 int*)l, 0, 0);
#else
  unsigned lds_off = (unsigned)(unsigned long long)(AS3 void*)l;
  asm volatile("global_load_async_to_lds_b32 %0, %1, off"
               :: "v"(lds_off), "v"(g)
               : "memory");
#endif
}

template <int N>
__device__ __forceinline__ void wait_async() {
#if __has_builtin(__builtin_amdgcn_s_wait_asynccnt)
  __builtin_amdgcn_s_wait_asynccnt(N);
#else
  asm volatile("s_wait_asynccnt %0" :: "i"(N) : "memory");
#endif
}

// ---- log-sum-exp helpers (hardware v_exp_f32 / v_log_f32) ----
__device__ __forceinline__ float lse2f(float a, float b) {
  float m = fmaxf(a, b);
  return m + __logf(__expf(a - m) + __expf(b - m));
}
__device__ __forceinline__ float lse3f(float a, float b, float c) {
  float m = fmaxf(a, fmaxf(b, c));
  return m + __logf(__expf(a - m) + __expf(b - m) + __expf(c - m));
}

// ---- wave32 shuffle reductions ----
__device__ __forceinline__ float wave_sum32(float v) {
  v += __shfl_xor(v, 16, 32);
  v += __shfl_xor(v, 8, 32);
  v += __shfl_xor(v, 4, 32);
  v += __shfl_xor(v, 2, 32);
  v += __shfl_xor(v, 1, 32);
  return v;
}
__device__ __forceinline__ float bfly_sum16(float v) {   // butterfly: all lanes get total
  v += __shfl_xor(v, 8, 16);
  v += __shfl_xor(v, 4, 16);
  v += __shfl_xor(v, 2, 16);
  v += __shfl_xor(v, 1, 16);
  return v;
}

// One workgroup (512 threads = 16 wave32s) per batch element.
// Thread j owns alpha[2j+1] (label j+1) and alpha[2j+2] (blank); alpha[0] is
// tracked redundantly per-thread. Two barriers per time step; async double-
// buffered row prefetch rides the same barriers.
extern "C" __global__ void __launch_bounds__(TKC)
forward_sum_loss_kernel(const float* __restrict__ attn,
                        const int* __restrict__ in_lens,
                        const int* __restrict__ out_lens,
                        float* __restrict__ out, int Bn, int TQ) {
  __shared__ float srow[2][TKC];   // double-buffered attn row (async targets)
  __shared__ float red[16];        // per-wave LSE partials
  __shared__ float shO[TKC];       // alpha at odd states (for left neighbor)
  __shared__ float shE[TKC];       // alpha at even states (for left neighbor)

  const int b    = blockIdx.x;
  const int j    = threadIdx.x;
  const int lane = j & 31;
  const int wid  = j >> 5;

  int kl = in_lens[b];
  kl = kl < 1 ? 1 : (kl > TKC ? TKC : kl);
  int ql = out_lens[b];
  ql = ql < 1 ? 1 : (ql > TQ ? TQ : ql);
  const bool valid = (j < kl);     // states 2j+1 and 2j+2 valid iff j < kl

  const float* base = attn + (size_t)b * (size_t)TQ * TKC;

  // ---- t = 0: prefetch rows 0 (and 1), init alphas ----
  async_row_load(base + j, &srow[0][j]);
  if (ql > 1) {
    async_row_load(base + TKC + j, &srow[1][j]);
    wait_async<1>();               // row 0 done (async loads complete in order)
  } else {
    wait_async<0>();
  }
  __syncthreads();

  float x = srow[0][j];            // padded column j+1 == attn[b, t, j]
  float p = valid ? __expf(x) : 0.0f;
  p = wave_sum32(p);
  if (lane == 0) red[wid] = p;
  __syncthreads();
  float lse = __logf(bfly_sum16(red[lane & 15]) + __expf(BLANK_LP));

  float a0 = BLANK_LP - lse;                 // alpha[0]   = logp blank
  float aO = (j == 0) ? (x - lse) : NEGV;    // alpha[1]   = logp label 1
  float aE = NEGV;                           // alpha[2j+2]
  shO[j] = aO;
  shE[j] = aE;

  // ---- sequential scan over query steps ----
  for (int t = 1; t < ql; ++t) {
    const bool pf = (t + 1 < ql);
    if (pf) async_row_load(base + (size_t)(t + 1) * TKC + j,
                           &srow[(t + 1) & 1][j]);
    if (pf) wait_async<1>(); else wait_async<0>();
    __syncthreads();                         // B0: row t + prev alphas visible

    x = srow[t & 1][j];
    const float aE_l = (j > 0) ? shE[j - 1] : NEGV;   // alpha[2j]   (prev t)
    const float aO_l = (j > 0) ? shO[j - 1] : NEGV;   // alpha[2j-1] (prev t)

    p = valid ? __expf(x) : 0.0f;
    p = wave_sum32(p);
    if (lane == 0) red[wid] = p;
    __syncthreads();                         // B1: partials ready, reads done

    lse = __logf(bfly_sum16(red[lane & 15]) + __expf(BLANK_LP));
    const float eb = BLANK_LP - lse;         // blank emission
    const float el = x - lse;                // label j+1 emission

    const float pm1 = (j == 0) ? a0 : aE_l;  // alpha[s-1]
    const float pm2 = (j > 0) ? aO_l : NEGV; // alpha[s-2]; skip illegal at s=1
    const float nO = lse3f(aO, pm1, pm2) + el;
    const float nE = lse2f(aE, aO) + eb;
    a0 += eb;                                // state 0: only self-loop
    aO = valid ? nO : NEGV;
    aE = valid ? nE : NEGV;
    shO[j] = aO;
    shE[j] = aE;
  }

  // ll = logaddexp(alpha[2kl], alpha[2kl-1]); both live in thread kl-1
  if (j == kl - 1) {
    const float ll = lse2f(aE, aO);
    const float loss = -ll / (float)kl;
    atomicAdd(out, loss / (float)Bn);        // mean over batch
  }
}

extern "C" void kernel_launch(void* const* d_in, const int* in_sizes, int n_in,
                              void* d_out, int out_size, void* d_ws, size_t ws_size,
                              hipStream_t stream) {
  (void)n_in; (void)out_size; (void)d_ws; (void)ws_size;
  const float* attn     = (const float*)d_in[0];
  const int*   in_lens  = (const int*)d_in[1];
  const int*   out_lens = (const int*)d_in[2];
  float*       out      = (float*)d_out;

  const int Bn = in_sizes[1];                       // batch (32)
  const int TQ = in_sizes[0] / (Bn * TKC);          // query length (2000)

  (void)hipMemsetAsync(d_out, 0, sizeof(float), stream);  // capture-safe zero init
  forward_sum_loss_kernel<<<Bn, TKC, 0, stream>>>(attn, in_lens, out_lens,
                                                  out, Bn, TQ);
}